// GCNNet_37460704756175
// MI455X (gfx1250) — compile-verified
//
#include <hip/hip_runtime.h>
#include <hip/hip_bf16.h>
#include <math.h>

// ---------------------------------------------------------------------------
// RGCN-ish link-prediction pipeline for MI455X (gfx1250, wave32).
//  - rel_emb @ W_{out,in} hoisted to 500-row GEMMs (WMMA f32 16x16x4)
//  - GraphConv projections as WMMA f32 GEMMs: 64-row M-blocks, K-tiled LDS
//    staged with global_load_async_to_lds_b128 (ASYNCcnt), one B fragment
//    shared by 4 stacked accumulators per wave (4x W-operand reuse).
//  - edge aggregation via L2-resident f32 atomics (the roofline bound here),
//    batched 4 edges/wave with global_prefetch of the next source row.
// ---------------------------------------------------------------------------

typedef float v2f __attribute__((ext_vector_type(2)));
typedef float v8f __attribute__((ext_vector_type(8)));

#define BM 64            // rows per block
#define KT 128           // K tile staged in LDS (all K here are multiples)
#define KP (KT + 4)      // padded row stride: 528B -> 16B aligned, bank-clean

__device__ __forceinline__ void store_tile(float* __restrict__ C, int Nc, int M,
                                           int row0, int col, int lane, v8f acc) {
    const int rbase = row0 + ((lane < 16) ? 0 : 8);
#pragma unroll
    for (int v = 0; v < 8; ++v) {
        int row = rbase + v;
        if (row < M) C[(size_t)row * Nc + col] = acc[v];
    }
}

// C[M x Nc] = A[M x K] @ W[K x Nc], all row-major f32.
// grid.x = ceil(M/64); blockDim.x = 32*(Nc/16). Wave w owns cols [16w,16w+16)
// and 4 M-subtiles; K is processed in KT-sized LDS-resident chunks.
__global__ __launch_bounds__(512)
void gemm_f32_wmma(const float* __restrict__ A, const float* __restrict__ W,
                   float* __restrict__ C, int M, int K, int Nc) {
    __shared__ __align__(16) float As[BM * KP];   // 33 KB

    const int tid  = threadIdx.x;
    const int row0 = blockIdx.x * BM;
    const int wave = tid >> 5;
    const int lane = tid & 31;
    const int col  = (wave << 4) + (lane & 15);   // this wave's column
    const int koff = (lane < 16) ? 0 : 2;         // ISA f32 A/B lane layout
    const int arow = lane & 15;

    v8f acc0 = {}, acc1 = {}, acc2 = {}, acc3 = {};

    const int CPR    = KT / 4;        // 16B chunks per staged row
    const int nchunk = BM * CPR;

    for (int k0 = 0; k0 < K; k0 += KT) {
        __syncthreads();              // previous chunk fully consumed

        // ---- stage A[row0:row0+BM, k0:k0+KT] into LDS via async DMA ----
        for (int c = tid; c < nchunk; c += blockDim.x) {
            int r  = c / CPR;
            int kc = c - r * CPR;
            int row = row0 + r;
            float* lp = As + r * KP + kc * 4;
            if (row < M) {
                const float* gp = A + (size_t)row * K + k0 + kc * 4;
                // generic LDS pointer: low 32 bits == LDS byte offset
                unsigned loff = (unsigned)(uintptr_t)lp;
                asm volatile("global_load_async_to_lds_b128 %0, %1, off"
                             :: "v"(loff), "v"(gp) : "memory");
            } else {
                lp[0] = 0.0f; lp[1] = 0.0f; lp[2] = 0.0f; lp[3] = 0.0f;
            }
        }
        asm volatile("s_wait_asynccnt 0x0" ::: "memory");
        __syncthreads();

        // ---- compute: one B fragment feeds 4 stacked WMMA tiles ----
        const float* __restrict__ Wc = W + col;
        const float* __restrict__ A0 = As + (arow +  0) * KP;
        const float* __restrict__ A1 = As + (arow + 16) * KP;
        const float* __restrict__ A2 = As + (arow + 32) * KP;
        const float* __restrict__ A3 = As + (arow + 48) * KP;
        for (int k = 0; k < KT; k += 4) {
            v2f b, a;
            b.x = Wc[(size_t)(k0 + k + koff) * Nc];
            b.y = Wc[(size_t)(k0 + k + koff + 1) * Nc];
            a.x = A0[k + koff]; a.y = A0[k + koff + 1];
            acc0 = __builtin_amdgcn_wmma_f32_16x16x4_f32(false, a, false, b,
                                                         (short)0, acc0, false, false);
            a.x = A1[k + koff]; a.y = A1[k + koff + 1];
            acc1 = __builtin_amdgcn_wmma_f32_16x16x4_f32(false, a, false, b,
                                                         (short)0, acc1, false, false);
            a.x = A2[k + koff]; a.y = A2[k + koff + 1];
            acc2 = __builtin_amdgcn_wmma_f32_16x16x4_f32(false, a, false, b,
                                                         (short)0, acc2, false, false);
            a.x = A3[k + koff]; a.y = A3[k + koff + 1];
            acc3 = __builtin_amdgcn_wmma_f32_16x16x4_f32(false, a, false, b,
                                                         (short)0, acc3, false, false);
        }
    }

    store_tile(C, Nc, M, row0 +  0, col, lane, acc0);
    store_tile(C, Nc, M, row0 + 16, col, lane, acc1);
    store_tile(C, Nc, M, row0 + 32, col, lane, acc2);
    store_tile(C, Nc, M, row0 + 48, col, lane, acc3);
}

__global__ void zerof(float* __restrict__ p, size_t n) {
    size_t i = (size_t)blockIdx.x * blockDim.x + threadIdx.x;
    if (i < n) p[i] = 0.0f;
}

__global__ void degrees_k(const int* __restrict__ esrc, const int* __restrict__ edst,
                          float* __restrict__ degO, float* __restrict__ degI, int E) {
    int e = blockIdx.x * blockDim.x + threadIdx.x;
    if (e < E) {
        atomicAdd(&degO[esrc[e]], 1.0f);
        atomicAdd(&degI[edst[e]], 1.0f);
    }
}

// One wave per edge: out_emb[src] += R_out[type]; in_emb[dst] += R_in[type].
__global__ void scatter_rel(const int* __restrict__ esrc, const int* __restrict__ edst,
                            const int* __restrict__ etype,
                            const float* __restrict__ Rout, const float* __restrict__ Rin,
                            float* __restrict__ out_emb, float* __restrict__ in_emb, int E) {
    int e = blockIdx.x * (blockDim.x >> 5) + (threadIdx.x >> 5);
    if (e >= E) return;
    int lane = threadIdx.x & 31;
    int s = esrc[e], d = edst[e], t = etype[e];
    const float* __restrict__ ro = Rout + (size_t)t * 128;
    const float* __restrict__ ri = Rin  + (size_t)t * 128;
    float* po = out_emb + (size_t)s * 128;
    float* pi = in_emb  + (size_t)d * 128;
#pragma unroll
    for (int j = lane; j < 128; j += 32) {
        atomicAdd(po + j, ro[j]);
        atomicAdd(pi + j, ri[j]);
    }
}

// xs = concat(feat, out_emb + deg_o*b_out, in_emb + deg_i*b_in) * rsqrt(max(deg_o,1))
__global__ void build_x(const float* __restrict__ feat,
                        const float* __restrict__ out_emb, const float* __restrict__ in_emb,
                        const float* __restrict__ b_out, const float* __restrict__ b_in,
                        const float* __restrict__ degO, const float* __restrict__ degI,
                        float* __restrict__ xs, float* __restrict__ odv, float* __restrict__ idgv) {
    int node = blockIdx.x;
    int j    = threadIdx.x;  // 0..127
    float dgo = degO[node], dgi = degI[node];
    float od  = rsqrtf(fmaxf(dgo, 1.0f));
    float idg = rsqrtf(fmaxf(dgi, 1.0f));
    if (j == 0) { odv[node] = od; idgv[node] = idg; }
    size_t nb = (size_t)node * 384, fb = (size_t)node * 128;
    xs[nb + j]       = feat[fb + j] * od;
    xs[nb + 128 + j] = (out_emb[fb + j] + dgo * b_out[j]) * od;
    xs[nb + 256 + j] = (in_emb[fb + j]  + dgi * b_in[j])  * od;
}

// 4 edges per wave: agg[dst] += hs[src]; prefetch next source row.
__global__ void scatter_feat(const int* __restrict__ esrc, const int* __restrict__ edst,
                             const float* __restrict__ hs, float* __restrict__ agg,
                             int E, int F) {
    int w    = blockIdx.x * (blockDim.x >> 5) + (threadIdx.x >> 5);
    int lane = threadIdx.x & 31;
    int e0   = w * 4;
#pragma unroll
    for (int q = 0; q < 4; ++q) {
        int e = e0 + q;
        if (e >= E) return;
        if (e + 1 < E) {
            const float* nxt = hs + (size_t)esrc[e + 1] * F;
            __builtin_prefetch(nxt + lane * (F >> 5), 0, 3);   // global_prefetch_b8
        }
        const float* __restrict__ src = hs + (size_t)esrc[e] * F;
        float* dst = agg + (size_t)edst[e] * F;
        for (int j = lane; j < F; j += 32) atomicAdd(dst + j, src[j]);
    }
}

// xs1 = relu(agg*idg + b) * od   (blockDim == F)
__global__ void relu_mid(const float* __restrict__ agg, const float* __restrict__ b,
                         const float* __restrict__ odv, const float* __restrict__ idgv,
                         float* __restrict__ xs1, int F) {
    int node = blockIdx.x, j = threadIdx.x;
    float v = agg[(size_t)node * F + j] * idgv[node] + b[j];
    v = fmaxf(v, 0.0f);
    xs1[(size_t)node * F + j] = v * odv[node];
}

// x2 = relu(agg*idg + b)   (blockDim == F)
__global__ void relu_final(const float* __restrict__ agg, const float* __restrict__ b,
                           const float* __restrict__ idgv, float* __restrict__ x2, int F) {
    int node = blockIdx.x, j = threadIdx.x;
    float v = agg[(size_t)node * F + j] * idgv[node] + b[j];
    x2[(size_t)node * F + j] = fmaxf(v, 0.0f);
}

// One wave per scored pair: ||h+r-t|| for pos and neg tails (shared h+r).
__global__ void score_k(const int* __restrict__ psrc, const int* __restrict__ pdst,
                        const int* __restrict__ ptype, const int* __restrict__ ndst,
                        const float* __restrict__ x2, const float* __restrict__ rel,
                        float* __restrict__ out, int EP) {
    int p = blockIdx.x * (blockDim.x >> 5) + (threadIdx.x >> 5);
    if (p >= EP) return;
    int lane = threadIdx.x & 31;
    const float* __restrict__ hs = x2  + (size_t)psrc[p]  * 128;
    const float* __restrict__ r  = rel + (size_t)ptype[p] * 128;
    const float* __restrict__ hd = x2  + (size_t)pdst[p]  * 128;
    const float* __restrict__ hn = x2  + (size_t)ndst[p]  * 128;
    float sp = 0.0f, sn = 0.0f;
#pragma unroll
    for (int j = lane; j < 128; j += 32) {
        float hr = hs[j] + r[j];
        float dp = hr - hd[j];
        float dn = hr - hn[j];
        sp += dp * dp;
        sn += dn * dn;
    }
#pragma unroll
    for (int off = 16; off > 0; off >>= 1) {
        sp += __shfl_xor(sp, off, 32);
        sn += __shfl_xor(sn, off, 32);
    }
    if (lane == 0) {
        out[p]      = sqrtf(sp);
        out[EP + p] = sqrtf(sn);
    }
}

static inline int ceildiv(int a, int b) { return (a + b - 1) / b; }

extern "C" void kernel_launch(void* const* d_in, const int* in_sizes, int n_in,
                              void* d_out, int out_size, void* d_ws, size_t ws_size,
                              hipStream_t stream) {
    const float* feat   = (const float*)d_in[0];
    const int*   esrc   = (const int*)  d_in[1];
    const int*   edst   = (const int*)  d_in[2];
    const int*   etype  = (const int*)  d_in[3];
    const int*   psrc   = (const int*)  d_in[4];
    const int*   pdst   = (const int*)  d_in[5];
    const int*   ptype  = (const int*)  d_in[6];
    /* d_in[7] = neg_src: unused by reference */
    const int*   ndst   = (const int*)  d_in[8];
    const float* rel    = (const float*)d_in[9];
    const float* W_out  = (const float*)d_in[10];
    const float* b_out  = (const float*)d_in[11];
    const float* W_in   = (const float*)d_in[12];
    const float* b_in   = (const float*)d_in[13];
    const float* W0     = (const float*)d_in[14];
    const float* b0     = (const float*)d_in[15];
    const float* W1     = (const float*)d_in[16];
    const float* b1     = (const float*)d_in[17];

    const int N  = in_sizes[0] / 128;
    const int E  = in_sizes[1];
    const int EP = in_sizes[4];
    const int R  = in_sizes[9] / 128;
    const int Rpad = ceildiv(R, 16) * 16;

    // ---- workspace layout (floats), with lifetime-based aliasing ----
    float* ws = (float*)d_ws;
    size_t o = 0;
    float* Rout    = ws + o; o += (size_t)Rpad * 128;
    float* Rin     = ws + o; o += (size_t)Rpad * 128;
    float* degO    = ws + o; o += (size_t)N;
    float* degI    = ws + o; o += (size_t)N;
    float* odv     = ws + o; o += (size_t)N;
    float* idgv    = ws + o; o += (size_t)N;
    float* out_emb = ws + o; o += (size_t)N * 128;
    float* in_emb  = ws + o; o += (size_t)N * 128;
    float* xs      = ws + o; o += (size_t)N * 384;
    float* hs0     = ws + o; o += (size_t)N * 256;
    // aliases (regions dead by the time they are reused):
    float* agg0 = xs;       // after hs0 GEMM, xs strip is dead
    float* xs1  = hs0;      // after scatter of hs0, hs0 is dead
    float* hs1  = xs;       // after relu_mid, agg0 is dead
    float* agg1 = in_emb;   // after build_x, in_emb is dead
    float* x2   = out_emb;  // after build_x, out_emb is dead

    // ---- 0: zero accumulators (degO..in_emb are contiguous) ----
    {
        size_t nz = (size_t)4 * N + (size_t)2 * N * 128;
        zerof<<<ceildiv((int)nz, 256), 256, 0, stream>>>(degO, nz);
    }

    // ---- 1: R_out = rel @ W_out ; R_in = rel @ W_in  (WMMA, tiny) ----
    gemm_f32_wmma<<<ceildiv(R, BM), 32 * (128 / 16), 0, stream>>>(rel, W_out, Rout, R, 128, 128);
    gemm_f32_wmma<<<ceildiv(R, BM), 32 * (128 / 16), 0, stream>>>(rel, W_in,  Rin,  R, 128, 128);

    // ---- 2: degrees + relation-embedding scatter ----
    degrees_k<<<ceildiv(E, 256), 256, 0, stream>>>(esrc, edst, degO, degI, E);
    scatter_rel<<<ceildiv(E, 8), 256, 0, stream>>>(esrc, edst, etype, Rout, Rin,
                                                   out_emb, in_emb, E);

    // ---- 3: assemble xs = concat(...) * od ; store od/idg ----
    build_x<<<N, 128, 0, stream>>>(feat, out_emb, in_emb, b_out, b_in,
                                   degO, degI, xs, odv, idgv);

    // ---- 4: hs0 = xs @ W0  [N,384]x[384,256] (WMMA) ----
    gemm_f32_wmma<<<ceildiv(N, BM), 32 * (256 / 16), 0, stream>>>(xs, W0, hs0, N, 384, 256);

    // ---- 5: agg0 = segsum(hs0[src] -> dst) ----
    {
        size_t nz = (size_t)N * 256;
        zerof<<<ceildiv((int)nz, 256), 256, 0, stream>>>(agg0, nz);
    }
    scatter_feat<<<ceildiv(E, 32), 256, 0, stream>>>(esrc, edst, hs0, agg0, E, 256);

    // ---- 6: xs1 = relu(agg0*idg + b0) * od ----
    relu_mid<<<N, 256, 0, stream>>>(agg0, b0, odv, idgv, xs1, 256);

    // ---- 7: hs1 = xs1 @ W1  [N,256]x[256,128] (WMMA) ----
    gemm_f32_wmma<<<ceildiv(N, BM), 32 * (128 / 16), 0, stream>>>(xs1, W1, hs1, N, 256, 128);

    // ---- 8: agg1 = segsum(hs1[src] -> dst) ----
    {
        size_t nz = (size_t)N * 128;
        zerof<<<ceildiv((int)nz, 256), 256, 0, stream>>>(agg1, nz);
    }
    scatter_feat<<<ceildiv(E, 32), 256, 0, stream>>>(esrc, edst, hs1, agg1, E, 128);

    // ---- 9: x2 = relu(agg1*idg + b1) ----
    relu_final<<<N, 128, 0, stream>>>(agg1, b1, idgv, x2, 128);

    // ---- 10: TransE scoring for pos/neg pairs ----
    score_k<<<ceildiv(EP, 8), 256, 0, stream>>>(psrc, pdst, ptype, ndst,
                                                x2, rel, (float*)d_out, EP);
}